// PatchConv3LayerClassifier_68547678044327
// MI455X (gfx1250) — compile-verified
//
#include <hip/hip_runtime.h>

typedef float v2f __attribute__((ext_vector_type(2)));
typedef float v8f __attribute__((ext_vector_type(8)));

#define NEG_SLOPE 0.01f

__device__ __forceinline__ void atomAddF(float* p, float v) {
    __hip_atomic_fetch_add(p, v, __ATOMIC_RELAXED, __HIP_MEMORY_SCOPE_AGENT);
}

// ---------------------------------------------------------------- degrees
__global__ __launch_bounds__(256) void k_deg(const int* __restrict__ src,
                                             const int* __restrict__ dst,
                                             float* __restrict__ dout,
                                             float* __restrict__ din, int nE) {
    int e = blockIdx.x * blockDim.x + threadIdx.x;
    if (e < nE) {
        atomAddF(dout + src[e], 1.0f);
        atomAddF(din + dst[e], 1.0f);
    }
}

__global__ __launch_bounds__(256) void k_rsqrt(float* __restrict__ d, int n) {
    int i = blockIdx.x * blockDim.x + threadIdx.x;
    if (i < n) d[i] = rsqrtf(fmaxf(d[i], 1.0f));
}

// ---------------------------------------------------------------- edge scatter
// AGG[dst] += X[src] * ew * iso[src]; D floats per edge, 4 per thread.
// For D=128 one wave == one edge, so force the edge index scalar: src/dst/ew
// become SMEM loads instead of 32 redundant vmem lane-loads.
template <int D>
__global__ __launch_bounds__(256) void k_scatter(const float* __restrict__ X,
                                                 float* __restrict__ AGG,
                                                 const int* __restrict__ src,
                                                 const int* __restrict__ dst,
                                                 const float* __restrict__ ew,
                                                 const float* __restrict__ iso,
                                                 int nE) {
    const int TPE = D / 4;  // threads per edge
    long long t = (long long)blockIdx.x * blockDim.x + threadIdx.x;
    if (t >= (long long)nE * TPE) return;
    int e = (int)(t / TPE);
    if (TPE == 32) e = __builtin_amdgcn_readfirstlane(e);  // wave-uniform
    int f = ((int)(t % TPE)) << 2;
    int s = src[e], d = dst[e];
    float w = ew[e] * iso[s];
    const float4 v = *(const float4*)(X + (size_t)s * D + f);
    float* o = AGG + (size_t)d * D + f;
    atomAddF(o + 0, v.x * w);
    atomAddF(o + 1, v.y * w);
    atomAddF(o + 2, v.z * w);
    atomAddF(o + 3, v.w * w);
}

// ---------------------------------------------------------------- WMMA fp32 GEMM
// C[M x NCOLS] = f( g(A[M x KDIM]) @ B[KDIM x NCOLS] )
//   A-read:  g(a) = A_RELU ? relu(a * isi[row]) : (A_ISI ? a * isi[row] : a)
//   C-write: f(v) = OUT_LEAKY ? leaky_relu(v) : v
// One wave computes a 16x16 tile via V_WMMA_F32_16X16X4_F32; 8 waves / block.
// The KDIM x 16 weight panel is staged once per block into LDS, interleaved as
// pairs {B[2p][n], B[2p+1][n]} so each step's B fragment is one ds_load_b64.
template <int KDIM, int NCOLS, int A_ISI, int A_RELU, int OUT_LEAKY>
__global__ __launch_bounds__(256) void k_gemm_wmma(const float* __restrict__ A,
                                                   const float* __restrict__ Bm,
                                                   float* __restrict__ C,
                                                   const float* __restrict__ isi,
                                                   int Mrows) {
    __shared__ float Bs[KDIM * 16];  // Bs[(k>>1)*32 + n*2 + (k&1)]
    const int tid   = threadIdx.x;
    const int ntile = blockIdx.y;

    // cooperative stage of the weight panel (coalesced 16-wide rows)
#pragma unroll
    for (int e = tid; e < KDIM * 16; e += 256) {
        int k = e >> 4, n = e & 15;
        Bs[(k >> 1) * 32 + n * 2 + (k & 1)] =
            Bm[(size_t)k * NCOLS + ntile * 16 + n];
    }
    __syncthreads();

    const int lane  = tid & 31;
    const int wave  = tid >> 5;
    const int mtile = blockIdx.x * 8 + wave;
    const int Mtiles = Mrows >> 4;
    if (mtile >= Mtiles) return;  // wave-uniform: EXEC stays all-ones for WMMA

    const int half = lane >> 4;   // 0: K={0,1}, 1: K={2,3} within each step
    const int l16  = lane & 15;
    const int arow = mtile * 16 + l16;   // A row this lane feeds (M = lane%16)

    float si = 1.0f;
    if (A_ISI) si = isi[arow];

    const float* Arow = A + (size_t)arow * KDIM + half * 2;
    const v2f*   Bp   = (const v2f*)Bs;  // Bp[p*16 + n] = {B[2p][n], B[2p+1][n]}

    v8f acc = {};
#pragma unroll 4
    for (int k = 0; k < KDIM; k += 4) {
        v2f a = *(const v2f*)(Arow + k);
        if (A_ISI) { a.x *= si; a.y *= si; }
        if (A_RELU) { a.x = fmaxf(a.x, 0.0f); a.y = fmaxf(a.y, 0.0f); }
        v2f b = Bp[((k >> 1) + half) * 16 + l16];  // ds_load_b64
        acc = __builtin_amdgcn_wmma_f32_16x16x4_f32(false, a, false, b,
                                                    (short)0, acc, false, false);
    }

    // D layout: VGPR j holds M = (half*8 + j), N = l16
    float* Crow = C + (size_t)(mtile * 16 + half * 8) * NCOLS + ntile * 16 + l16;
#pragma unroll
    for (int j = 0; j < 8; j++) {
        float v = acc[j];
        if (OUT_LEAKY) v = (v > 0.0f) ? v : NEG_SLOPE * v;
        Crow[(size_t)j * NCOLS] = v;
    }
}

// ---------------------------------------------------------------- mean pool
// pooled[g][f] = mean_n relu(AGG[g*npg+n][f] * isi[..]) ; HID=128, blockDim=128
__global__ __launch_bounds__(128) void k_pool(const float* __restrict__ AGG,
                                              const float* __restrict__ isi,
                                              float* __restrict__ pooled,
                                              int npg) {
    int g = blockIdx.x;
    int f = threadIdx.x;  // 0..127
    float s = 0.0f;
    int base = g * npg;
    for (int i = 0; i < npg; i++) {
        int n = base + i;
        float v = AGG[(size_t)n * 128 + f] * isi[n];
        s += fmaxf(v, 0.0f);
    }
    pooled[(size_t)g * 128 + f] = s / (float)npg;
}

// ---------------------------------------------------------------- head
__global__ __launch_bounds__(64) void k_head1(const float* __restrict__ pooled,
                                              const float* __restrict__ Wl,
                                              float* __restrict__ tmp) {
    int g = blockIdx.x, j = threadIdx.x;  // 64
    float s = 0.0f;
    for (int k = 0; k < 128; k++) s += pooled[g * 128 + k] * Wl[k * 64 + j];
    tmp[g * 64 + j] = s;
}

__global__ __launch_bounds__(16) void k_head2(const float* __restrict__ tmp,
                                              const float* __restrict__ Wc,
                                              float* __restrict__ out) {
    int g = blockIdx.x, j = threadIdx.x;  // 16
    float s = 0.0f;
    for (int k = 0; k < 64; k++) s += tmp[g * 64 + k] * Wc[k * 16 + j];
    out[g * 16 + j] = s;
}

// ---------------------------------------------------------------- launch
extern "C" void kernel_launch(void* const* d_in, const int* in_sizes, int n_in,
                              void* d_out, int out_size, void* d_ws, size_t ws_size,
                              hipStream_t stream) {
    const float* node_feats = (const float*)d_in[0];  // [N,64]
    const float* ew         = (const float*)d_in[1];  // [E]
    const int*   esrc       = (const int*)d_in[2];    // [E]
    const int*   edst       = (const int*)d_in[3];    // [E]
    const float* W1         = (const float*)d_in[4];  // [64,256]
    const float* W2         = (const float*)d_in[5];  // [256,128]
    const float* W3         = (const float*)d_in[6];  // [128,128]
    const float* Wl         = (const float*)d_in[7];  // [128,64]
    const float* Wc         = (const float*)d_in[8];  // [64,16]
    float* out = (float*)d_out;

    const int N = in_sizes[0] / 64;
    const int E = in_sizes[1];
    const int B = out_size / 16;
    const int npg = N / B;

    // workspace layout (floats)
    float* ws   = (float*)d_ws;
    float* iso  = ws;                              // [N]
    float* isi  = iso + N;                         // [N]
    float* agg1 = isi + N;                         // [N,64]
    float* h1   = agg1 + (size_t)N * 64;           // [N,256]
    float* xw2  = h1 + (size_t)N * 256;            // [N,128]
    float* agg2 = xw2 + (size_t)N * 128;           // [N,128]
    float* pooled = agg2 + (size_t)N * 128;        // [B,128]
    float* tmp    = pooled + (size_t)B * 128;      // [B,64]
    float* xw3  = h1;   // h1 dead after GEMM2
    float* agg3 = xw2;  // xw2 dead after scatter2

    const int T = 256;

    // degrees -> iso/isi
    hipMemsetAsync(iso, 0, sizeof(float) * 2 * N, stream);
    k_deg<<<(E + T - 1) / T, T, 0, stream>>>(esrc, edst, iso, isi, E);
    k_rsqrt<<<(2 * N + T - 1) / T, T, 0, stream>>>(iso, 2 * N);

    // layer 1: aggregate(64) -> leaky((agg*isi) @ W1)
    hipMemsetAsync(agg1, 0, sizeof(float) * (size_t)N * 64, stream);
    {
        long long tot = (long long)E * 16;
        k_scatter<64><<<(unsigned)((tot + T - 1) / T), T, 0, stream>>>(
            node_feats, agg1, esrc, edst, ew, iso, E);
    }
    {
        dim3 g((N / 16 + 7) / 8, 256 / 16);
        k_gemm_wmma<64, 256, 1, 0, 1><<<g, T, 0, stream>>>(agg1, W1, h1, isi, N);
    }

    // layer 2: xw2 = h1 @ W2 ; aggregate(128)
    {
        dim3 g((N / 16 + 7) / 8, 128 / 16);
        k_gemm_wmma<256, 128, 0, 0, 0><<<g, T, 0, stream>>>(h1, W2, xw2, isi, N);
    }
    hipMemsetAsync(agg2, 0, sizeof(float) * (size_t)N * 128, stream);
    {
        long long tot = (long long)E * 32;
        k_scatter<128><<<(unsigned)((tot + T - 1) / T), T, 0, stream>>>(
            xw2, agg2, esrc, edst, ew, iso, E);
    }

    // layer 3: xw3 = relu(agg2*isi) @ W3 ; aggregate(128)
    {
        dim3 g((N / 16 + 7) / 8, 128 / 16);
        k_gemm_wmma<128, 128, 1, 1, 0><<<g, T, 0, stream>>>(agg2, W3, xw3, isi, N);
    }
    hipMemsetAsync(agg3, 0, sizeof(float) * (size_t)N * 128, stream);
    {
        long long tot = (long long)E * 32;
        k_scatter<128><<<(unsigned)((tot + T - 1) / T), T, 0, stream>>>(
            xw3, agg3, esrc, edst, ew, iso, E);
    }

    // pool + head
    k_pool<<<B, 128, 0, stream>>>(agg3, isi, pooled, npg);
    k_head1<<<B, 64, 0, stream>>>(pooled, Wl, tmp);
    k_head2<<<B, 16, 0, stream>>>(tmp, Wc, out);
}